// xLSTMBlock_41042707480757
// MI455X (gfx1250) — compile-verified
//
#include <hip/hip_runtime.h>
#include <hip/hip_bf16.h>
#include <stdint.h>

#define HID 1024
#define NH 16
#define DH 64
#define IN_DIM 1024
#define OUT_DIM 4128   // 3*HID + 2*NH + HID
#define BATCH 4
#define SEQ 1024
#define MTOT (BATCH * SEQ)   // 4096
#define XEPS 1e-6f

typedef float v2f __attribute__((ext_vector_type(2)));
typedef float v8f __attribute__((ext_vector_type(8)));
typedef unsigned int u32x4 __attribute__((ext_vector_type(4)));
typedef int i32x8 __attribute__((ext_vector_type(8)));
typedef int i32x4 __attribute__((ext_vector_type(4)));

// ---------------------------------------------------------------------------
// Tensor Data Mover helpers
// ---------------------------------------------------------------------------
__device__ __forceinline__ unsigned lds_off(const void* p) {
    // Flat LDS addresses carry the aperture in the high 32 bits; low 32 bits
    // are the wave-relative LDS byte offset (ISA 10.2 aperture mapping).
    return (unsigned)(uintptr_t)p;
}

// DMA a (rows=64) x (cols=32 f32) tile from a row-major matrix with row
// stride IN_DIM floats into LDS, with hardware padding of 2 DWORDs after
// every 32 DWORDs (=> LDS row stride 34 floats). OOB rows/cols (beyond
// rows_rem / k_rem from the tile origin) are zero-filled by the TDM.
__device__ __forceinline__ void tdm_load_tile(unsigned lds_addr,
                                              const float* gtile,
                                              unsigned rows_rem,
                                              unsigned k_rem) {
    unsigned long long ga = (unsigned long long)(uintptr_t)gtile;

    u32x4 g0;
    g0[0] = 1u;                                             // count=1 (valid)
    g0[1] = lds_addr;                                       // lds_addr
    g0[2] = (unsigned)ga;                                   // global_addr[31:0]
    g0[3] = (unsigned)((ga >> 32) & 0x01FFFFFFu)            // global_addr[56:32]
          | (2u << 30);                                     // type=2 ("image")

    i32x8 g1;
    g1[0] = (int)((2u << 16)      // data_size = 4 bytes
                | (1u << 20)      // pad_enable
                | (4u << 22)      // pad_interval: 32 DWORDs
                | (1u << 25));    // pad_amount:   2 DWORDs
    g1[1] = (int)((k_rem & 0xFFFFu) << 16);                 // tensor_dim0[15:0]
    g1[2] = (int)((k_rem >> 16) | ((rows_rem & 0xFFFFu) << 16)); // td0 hi | td1 lo
    g1[3] = (int)((rows_rem >> 16) | (32u << 16));          // td1 hi | tile_dim0=32
    g1[4] = 64;                                             // tile_dim1=64 (tile_dim2=0)
    g1[5] = (int)IN_DIM;                                    // tensor_dim0_stride lo
    g1[6] = 0;                                              // stride0 hi | stride1 lo
    g1[7] = 0;                                              // stride1 hi

    i32x4 z4 = {0, 0, 0, 0};                                // groups 2/3: 2-D tensor
    i32x8 z8 = {0, 0, 0, 0, 0, 0, 0, 0};                    // clang-23 6-arg form
    __builtin_amdgcn_tensor_load_to_lds(g0, g1, z4, z4, z8, 0);
}

// ---------------------------------------------------------------------------
// Phase 1: proj[m][n] = sum_k x[m][k] * W[n][k] + b[n]
// f32 WMMA 16x16x4. Block = 128 threads (4 waves), 64x64 tile, each wave 32x32.
// TDM double-buffered LDS staging (no VGPR round-trip).
// ---------------------------------------------------------------------------
#define KC 32
#define LDA 34            // LDS row stride in floats (32 data + 2 pad)
#define ASZ (64 * LDA)    // floats per staged tile

__global__ __launch_bounds__(128)
void xlstm_proj_gemm(const float* __restrict__ X,
                     const float* __restrict__ W,
                     const float* __restrict__ bias,
                     float* __restrict__ proj) {
    __shared__ __align__(16) float smem[4 * ASZ];   // A0, A1, B0, B1

    const int m0   = blockIdx.x * 64;
    const int n0   = blockIdx.y * 64;
    const int tid  = threadIdx.x;
    const int wave = tid >> 5;
    const int lane = tid & 31;
    const int wm   = (wave >> 1) * 32;   // wave's M offset inside tile
    const int wn   = (wave & 1) * 32;    // wave's N offset inside tile
    const int l16   = lane & 15;
    const int lhalf = lane >> 4;         // 0: K pair {0,1}; 1: K pair {2,3}

    const unsigned rowsA = (unsigned)(MTOT - m0);
    const unsigned rowsB = (unsigned)(OUT_DIM - n0);   // ragged edge: 32 at last block
    const bool issuer = (wave == 0);

    v8f acc[2][2] = {};

    // ---- prime the pipeline: chunk 0 into buffer 0 ----
    if (issuer) {
        tdm_load_tile(lds_off(&smem[0]),        X + (size_t)m0 * IN_DIM, rowsA, IN_DIM);
        tdm_load_tile(lds_off(&smem[2 * ASZ]),  W + (size_t)n0 * IN_DIM, rowsB, IN_DIM);
    }

    const int NCH = IN_DIM / KC;   // 32 chunks
    for (int c = 0; c < NCH; ++c) {
        const int buf = c & 1;

        if (issuer) {
            if (c + 1 < NCH) {
                const int kk = (c + 1) * KC;
                tdm_load_tile(lds_off(&smem[(buf ^ 1) * ASZ]),
                              X + (size_t)m0 * IN_DIM + kk, rowsA, (unsigned)(IN_DIM - kk));
                tdm_load_tile(lds_off(&smem[(2 + (buf ^ 1)) * ASZ]),
                              W + (size_t)n0 * IN_DIM + kk, rowsB, (unsigned)(IN_DIM - kk));
                __builtin_amdgcn_s_wait_tensorcnt(2);  // chunk c's two DMAs done
            } else {
                __builtin_amdgcn_s_wait_tensorcnt(0);
            }
        }
        __syncthreads();

        const float* As = &smem[buf * ASZ];
        const float* Bs = &smem[(2 + buf) * ASZ];

        #pragma unroll
        for (int t = 0; t < KC; t += 4) {
            const int kcol = t + 2 * lhalf;
            v2f afrag[2], bfrag[2];
            #pragma unroll
            for (int a = 0; a < 2; ++a)
                afrag[a] = *(const v2f*)&As[(wm + 16 * a + l16) * LDA + kcol];
            #pragma unroll
            for (int b = 0; b < 2; ++b)
                bfrag[b] = *(const v2f*)&Bs[(wn + 16 * b + l16) * LDA + kcol];
            #pragma unroll
            for (int a = 0; a < 2; ++a)
                #pragma unroll
                for (int b = 0; b < 2; ++b)
                    acc[a][b] = __builtin_amdgcn_wmma_f32_16x16x4_f32(
                        false, afrag[a], false, bfrag[b],
                        (short)0, acc[a][b], false, false);
        }
        __syncthreads();   // all waves done with buf before TDM overwrites it
    }

    // ---- epilogue: add bias, store ----
    #pragma unroll
    for (int a = 0; a < 2; ++a) {
        #pragma unroll
        for (int b = 0; b < 2; ++b) {
            const int mbase = m0 + wm + 16 * a;
            const int nn    = n0 + wn + 16 * b + l16;
            if (nn < OUT_DIM) {
                const float bi = bias[nn];
                #pragma unroll
                for (int r = 0; r < 8; ++r) {
                    const int m = mbase + r + 8 * lhalf;  // C layout: VGPR r = row r / r+8
                    proj[(size_t)m * OUT_DIM + nn] = acc[a][b][r] + bi;
                }
            }
        }
    }
}

// ---------------------------------------------------------------------------
// Phase 2: sequential mLSTM scan. One block per (batch, head) cell.
// 256 threads: thread = (row r = tid&63, column quadrant cq = tid>>6).
// C state: 16 registers / thread.
// ---------------------------------------------------------------------------
__global__ __launch_bounds__(256)
void xlstm_scan(const float* __restrict__ proj,
                float* __restrict__ out,
                float* __restrict__ Cf,
                float* __restrict__ nf,
                float* __restrict__ mf) {
    const int bh = blockIdx.x;      // 0..63
    const int b  = bh >> 4;
    const int h  = bh & 15;
    const int tid = threadIdx.x;
    const int r   = tid & 63;
    const int cq  = tid >> 6;       // 0..3
    const int c0  = cq * 16;

    __shared__ float kbuf[64], qbuf[64], vbuf[64], obuf[64];
    __shared__ float scal[2];       // [0]=i', [1]=f'
    __shared__ float num4[4][64];
    __shared__ float den2[2];

    float Creg[16];
    #pragma unroll
    for (int j = 0; j < 16; ++j) Creg[j] = 0.f;
    float nreg = 0.f;               // valid for tid < 64
    float mreg = 0.f;               // valid for tid == 0

    const size_t projBase = (size_t)b * SEQ * OUT_DIM;
    const int qo  = h * DH;
    const int ko  = HID + h * DH;
    const int vo  = 2 * HID + h * DH;
    const int igo = 3 * HID + h;
    const int fgo = 3 * HID + NH + h;
    const int ogo = 3 * HID + 2 * NH + h * DH;

    for (int s = 0; s < SEQ; ++s) {
        const float* p = proj + projBase + (size_t)s * OUT_DIM;

        if (tid < 64) {
            qbuf[tid] = p[qo + tid];
            kbuf[tid] = p[ko + tid];
            vbuf[tid] = p[vo + tid];
            float og  = p[ogo + tid];
            obuf[tid] = 1.f / (1.f + __expf(-og));
        }
        if (tid == 0) {
            float ig = p[igo];
            float fg = p[fgo];
            float m_t = fmaxf(fg + mreg, ig);
            scal[0] = __expf(ig - m_t);
            scal[1] = __expf(fg + mreg - m_t);
            mreg = m_t;
        }
        __syncthreads();

        const float i_p = scal[0];
        const float f_p = scal[1];
        const float vr  = vbuf[r];
        float numpart = 0.f;
        #pragma unroll
        for (int j = 0; j < 16; ++j) {
            const float kj = kbuf[c0 + j];
            Creg[j] = f_p * Creg[j] + i_p * vr * kj;
            numpart += Creg[j] * qbuf[c0 + j];
        }
        num4[cq][r] = numpart;

        if (tid < 64) {
            nreg = f_p * nreg + i_p * kbuf[r];
            float dp = nreg * qbuf[r];
            #pragma unroll
            for (int off = 16; off; off >>= 1)
                dp += __shfl_xor(dp, off, 32);
            if ((tid & 31) == 0) den2[tid >> 5] = dp;
        }
        __syncthreads();

        if (tid < 64) {
            const float num = num4[0][r] + num4[1][r] + num4[2][r] + num4[3][r];
            const float den = fabsf(den2[0] + den2[1]);
            const float ht  = obuf[r] * (num / (den + XEPS));
            out[((size_t)b * SEQ + s) * HID + h * DH + r] = ht;
        }
        __syncthreads();   // protect LDS buffers before next stage
    }

    // ---- final states ----
    #pragma unroll
    for (int j = 0; j < 16; ++j)
        Cf[(((size_t)bh) * DH + r) * DH + c0 + j] = Creg[j];
    if (tid < 64) nf[(size_t)bh * DH + r] = nreg;
    if (tid == 0) mf[bh] = mreg;
}

// ---------------------------------------------------------------------------
extern "C" void kernel_launch(void* const* d_in, const int* in_sizes, int n_in,
                              void* d_out, int out_size, void* d_ws, size_t ws_size,
                              hipStream_t stream) {
    const float* x    = (const float*)d_in[0];
    const float* W    = (const float*)d_in[1];
    const float* bias = (const float*)d_in[2];

    float* out = (float*)d_out;
    float* Cf  = out + (size_t)BATCH * SEQ * HID;          // 4,194,304
    float* nf  = Cf  + (size_t)BATCH * NH * DH * DH;       // +262,144
    float* mf  = nf  + (size_t)BATCH * NH * DH;            // +4,096

    float* proj = (float*)d_ws;   // 4096 x 4128 f32 = 67.6 MB scratch

    dim3 gemmGrid(MTOT / 64, (OUT_DIM + 63) / 64);         // 64 x 65
    xlstm_proj_gemm<<<gemmGrid, 128, 0, stream>>>(x, W, bias, proj);

    xlstm_scan<<<BATCH * NH, 256, 0, stream>>>(proj, out, Cf, nf, mf);
}